// DoubleResNetUpDown_35545149342152
// MI455X (gfx1250) — compile-verified
//
#include <hip/hip_runtime.h>
#include <hip/hip_bf16.h>

// DoubleResNetUpDown: per-row chain of 30 tiny residual blocks on a 6-float
// state.  VALU-issue-bound on MI455X; 4 rows/thread for VOPD pairing,
// wave-uniform weights via scalar loads software-pipelined one layer ahead,
// inputs staged to LDS with gfx1250 async global->LDS loads
// (GLOBAL_LOAD_ASYNC_TO_LDS_B128 + s_wait_asynccnt).

#define NLAYERS 30
#define FD 5
#define TPB 256
#define ROWS 4
#define TILE (TPB * ROWS)   // 1024 rows per block

#define AS1 __attribute__((address_space(1)))
#define AS3 __attribute__((address_space(3)))

typedef int v4i __attribute__((ext_vector_type(4)));

#if defined(__HIP_DEVICE_COMPILE__) && defined(__gfx1250__) && \
    __has_builtin(__builtin_amdgcn_global_load_async_to_lds_b128) && \
    __has_builtin(__builtin_amdgcn_s_wait_asynccnt)
#define USE_ASYNC_LDS 1
#else
#define USE_ASYNC_LDS 0
#endif

__global__ __launch_bounds__(TPB) void
double_resnet_updown_kernel(const float* __restrict__ gx1,
                            const float* __restrict__ gx2,
                            const float* __restrict__ W1,
                            const float* __restrict__ B1,
                            const float* __restrict__ W2,
                            float* __restrict__ y1,
                            float* __restrict__ y2,
                            int nrows)
{
    const int tid      = threadIdx.x;
    const int tileBase = blockIdx.x * TILE;

    float c1[ROWS];
    float c2[ROWS][FD];

    const bool fullTile = (tileBase + TILE) <= nrows;

#if USE_ASYNC_LDS
    __shared__ __align__(16) float lx1[TILE];        //  4 KB
    __shared__ __align__(16) float lx2[TILE * FD];   // 20 KB
    if (fullTile) {
        const float* gsrc1 = gx1 + (size_t)tileBase;
        const float* gsrc2 = gx2 + (size_t)tileBase * FD;

        // x2 tile: 5120 words = 1280 x b128, 5 per thread, 16B-aligned.
        #pragma unroll
        for (int k = 0; k < (TILE * FD) / (TPB * 4); ++k) {
            const int off = (tid + k * TPB) * 4;  // word offset
            __builtin_amdgcn_global_load_async_to_lds_b128(
                (AS1 v4i*)(gsrc2 + off), (AS3 v4i*)(lx2 + off), 0, 0);
        }
        // x1 tile: 1024 words = 256 x b128, 1 per thread.
        {
            const int off = tid * 4;
            __builtin_amdgcn_global_load_async_to_lds_b128(
                (AS1 v4i*)(gsrc1 + off), (AS3 v4i*)(lx1 + off), 0, 0);
        }
        __builtin_amdgcn_s_wait_asynccnt(0);
        __syncthreads();

        #pragma unroll
        for (int i = 0; i < ROWS; ++i) {
            const int r = tid + i * TPB;
            c1[i] = lx1[r];
            #pragma unroll
            for (int j = 0; j < FD; ++j) c2[i][j] = lx2[r * FD + j];
        }
    } else
#endif
    {
        #pragma unroll
        for (int i = 0; i < ROWS; ++i) {
            const int r  = tileBase + tid + i * TPB;
            const bool ok = (r < nrows);
            c1[i] = ok ? gx1[r] : 0.0f;
            #pragma unroll
            for (int j = 0; j < FD; ++j)
                c2[i][j] = ok ? gx2[(size_t)r * FD + j] : 0.0f;
        }
    }

    // 30 residual blocks.  Weights are wave-uniform -> SMEM scalar loads,
    // software-pipelined one layer ahead so the s_load latency overlaps the
    // ~68-VALU-op layer body (the prefetch copies are s_movs that co-issue
    // with the VALU stream).
    float pw1[FD], pw2[FD], pb;
    #pragma unroll
    for (int j = 0; j < FD; ++j) {
        pw1[j] = W1[j];
        pw2[j] = W2[j];
    }
    pb = B1[0];

    for (int l = 0; l < NLAYERS; ++l) {
        float w1[FD], w2[FD];
        const float bb = pb;
        #pragma unroll
        for (int j = 0; j < FD; ++j) {
            w1[j] = pw1[j];
            w2[j] = pw2[j];
        }
        if (l + 1 < NLAYERS) {
            #pragma unroll
            for (int j = 0; j < FD; ++j) {
                pw1[j] = W1[(l + 1) * FD + j];
                pw2[j] = W2[(l + 1) * FD + j];
            }
            pb = B1[l + 1];
        }

        #pragma unroll
        for (int i = 0; i < ROWS; ++i) {
            // c1 += relu(c2) . w1 + b1
            float acc = c1[i] + bb;
            #pragma unroll
            for (int j = 0; j < FD; ++j)
                acc = fmaf(fmaxf(c2[i][j], 0.0f), w1[j], acc);
            c1[i] = acc;
            // c2 += relu(c1) * w2
            const float r1 = fmaxf(acc, 0.0f);
            #pragma unroll
            for (int j = 0; j < FD; ++j)
                c2[i][j] = fmaf(r1, w2[j], c2[i][j]);
        }
    }

    #pragma unroll
    for (int i = 0; i < ROWS; ++i) {
        const int r = tileBase + tid + i * TPB;
        if (r < nrows) {
            y1[r] = c1[i];
            #pragma unroll
            for (int j = 0; j < FD; ++j)
                y2[(size_t)r * FD + j] = c2[i][j];
        }
    }
}

extern "C" void kernel_launch(void* const* d_in, const int* in_sizes, int n_in,
                              void* d_out, int out_size, void* d_ws, size_t ws_size,
                              hipStream_t stream) {
    const float* x1 = (const float*)d_in[0];   // (B,1)
    const float* x2 = (const float*)d_in[1];   // (B,F)
    const float* W1 = (const float*)d_in[2];   // (L,1,F)
    const float* b1 = (const float*)d_in[3];   // (L,1)
    const float* W2 = (const float*)d_in[4];   // (L,F,1)

    const int nrows = in_sizes[0];             // B
    float* y1 = (float*)d_out;                 // (B,)   first output
    float* y2 = y1 + nrows;                    // (B,F)  second output

    const int grid = (nrows + TILE - 1) / TILE;
    double_resnet_updown_kernel<<<grid, TPB, 0, stream>>>(
        x1, x2, W1, b1, W2, y1, y2, nrows);
    (void)d_ws; (void)ws_size; (void)n_in; (void)out_size;
}